// QuantizerEMA_6150393168137
// MI455X (gfx1250) — compile-verified
//
#include <hip/hip_runtime.h>

typedef __attribute__((ext_vector_type(2))) float v2f;
typedef __attribute__((ext_vector_type(8))) float v8f;

#define KCLUST 1024
#define DDIM   32
#define DECAYF 0.99f
#define OMDF   0.01f     // 1 - DECAY
#define COMMITF 0.1f
#define EPSF   1e-5f

// workspace layout (float elements)
#define WS_COUNTS 0          // 1024
#define WS_DW     1024       // 32768
#define WS_LOSS   33792      // 1
#define WS_E2     33800      // 1024
#define WS_ZERO_N 33793

// ---------------------------------------------------------------------------
// Init: zero accumulators, precompute ||e_k||^2
// ---------------------------------------------------------------------------
__global__ __launch_bounds__(256)
void vq_init_kernel(float* __restrict__ ws, const float* __restrict__ emb) {
  int tid = blockIdx.x * blockDim.x + threadIdx.x;
  if (tid < WS_ZERO_N) ws[tid] = 0.0f;
  if (tid < KCLUST) {
    float s = 0.0f;
#pragma unroll
    for (int d = 0; d < DDIM; ++d) {
      float e = emb[tid * DDIM + d];
      s = fmaf(e, e, s);
    }
    ws[WS_E2 + tid] = s;
  }
}

// ---------------------------------------------------------------------------
// helpers
// ---------------------------------------------------------------------------
__device__ __forceinline__ void load_bfrag(v2f (&b)[8], const float* brow) {
#pragma unroll
  for (int s = 0; s < 8; ++s) b[s] = *(const v2f*)(brow + 4 * s);
}

__device__ __forceinline__ void tile_step(const v2f (&a0)[8], const v2f (&a1)[8],
                                          const v2f (&b)[8], float e2v, int n,
                                          float (&bestd0)[8], int (&besti0)[8],
                                          float (&bestd1)[8], int (&besti1)[8]) {
  v8f c0 = {0.f, 0.f, 0.f, 0.f, 0.f, 0.f, 0.f, 0.f};
  v8f c1 = {0.f, 0.f, 0.f, 0.f, 0.f, 0.f, 0.f, 0.f};
#pragma unroll
  for (int s = 0; s < 8; ++s) {
    c0 = __builtin_amdgcn_wmma_f32_16x16x4_f32(false, a0[s], false, b[s],
                                               (short)0, c0, false, false);
    c1 = __builtin_amdgcn_wmma_f32_16x16x4_f32(false, a1[s], false, b[s],
                                               (short)0, c1, false, false);
  }
  // C layout: VGPR v, lanes 0-15 -> row M=v; lanes 16-31 -> row M=8+v
#pragma unroll
  for (int v = 0; v < 8; ++v) {
    float sc0 = fmaf(-2.0f, c0[v], e2v);   // ||e||^2 - 2 z.e
    if (sc0 < bestd0[v]) { bestd0[v] = sc0; besti0[v] = n; }
    float sc1 = fmaf(-2.0f, c1[v], e2v);
    if (sc1 < bestd1[v]) { bestd1[v] = sc1; besti1[v] = n; }
  }
}

// ---------------------------------------------------------------------------
// Main: per-wave 32-row tile (2x M register blocking), double-buffered B;
// WMMA f32 16x16x4 distance GEMM + argmin + outputs + EMA scatter + loss
// ---------------------------------------------------------------------------
__global__ __launch_bounds__(256)
void vq_main_kernel(const float* __restrict__ zf,
                    const float* __restrict__ emb,
                    float* __restrict__ ws,
                    float* __restrict__ outQ,
                    float* __restrict__ outIdx,
                    int numRowTiles) {
  __shared__ float s_e2[KCLUST];
  const int tid = threadIdx.x;
  for (int i = tid; i < KCLUST; i += 256) s_e2[i] = ws[WS_E2 + i];
  __syncthreads();

  const int wave   = tid >> 5;
  const int lane   = tid & 31;
  const int lane15 = lane & 15;
  const int half   = lane >> 4;

  const int rowTile = blockIdx.x * 8 + wave;   // 32-row tiles
  if (rowTile >= numRowTiles) return;
  const int r0 = rowTile * 32;

  // A fragments: two 16x32 f32 tiles of z.
  // ISA 16x4 f32 A layout: lanes 0-15 -> (K, K+1), lanes 16-31 -> (K+2, K+3)
  v2f a0[8], a1[8];
  {
    const float* zrow0 = zf + (size_t)(r0 + lane15) * DDIM + 2 * half;
    const float* zrow1 = zf + (size_t)(r0 + 16 + lane15) * DDIM + 2 * half;
#pragma unroll
    for (int s = 0; s < 8; ++s) {
      a0[s] = *(const v2f*)(zrow0 + 4 * s);
      a1[s] = *(const v2f*)(zrow1 + 4 * s);
    }
  }

  float bestd0[8], bestd1[8];
  int   besti0[8], besti1[8];
#pragma unroll
  for (int v = 0; v < 8; ++v) {
    bestd0[v] = 3.0e38f; besti0[v] = 0;
    bestd1[v] = 3.0e38f; besti1[v] = 0;
  }

  // B base for this lane: B[k][n] = E[n][k]; tile t starts at row t*16.
  const float* embBase = emb + (size_t)lane15 * DDIM + 2 * half;

  v2f b0[8], b1[8];
  load_bfrag(b0, embBase);                       // tile 0

  // 64 tiles of 16 embeddings, two per iteration, double-buffered.
#pragma unroll 1
  for (int tp = 0; tp < KCLUST / 32; ++tp) {
    const int t0 = 2 * tp;
    const int t1 = t0 + 1;

    load_bfrag(b1, embBase + (size_t)t1 * 16 * DDIM);   // prefetch tile t1
    tile_step(a0, a1, b0, s_e2[t0 * 16 + lane15], t0 * 16 + lane15,
              bestd0, besti0, bestd1, besti1);

    const int tn = (tp < KCLUST / 32 - 1) ? (t0 + 2) : 0;  // wrap: harmless
    load_bfrag(b0, embBase + (size_t)tn * 16 * DDIM);   // prefetch tile t0+2
    tile_step(a0, a1, b1, s_e2[t1 * 16 + lane15], t1 * 16 + lane15,
              bestd0, besti0, bestd1, besti1);
  }

  // Reduce argmin across the 16 lanes of each half (masks <= 8 stay in-half).
#pragma unroll
  for (int v = 0; v < 8; ++v) {
#pragma unroll
    for (int m = 8; m >= 1; m >>= 1) {
      float od0 = __shfl_xor(bestd0[v], m, 32);
      int   oi0 = __shfl_xor(besti0[v], m, 32);
      if (od0 < bestd0[v] || (od0 == bestd0[v] && oi0 < besti0[v])) {
        bestd0[v] = od0; besti0[v] = oi0;
      }
      float od1 = __shfl_xor(bestd1[v], m, 32);
      int   oi1 = __shfl_xor(besti1[v], m, 32);
      if (od1 < bestd1[v] || (od1 == bestd1[v] && oi1 < besti1[v])) {
        bestd1[v] = od1; besti1[v] = oi1;
      }
    }
  }

  float* counts = ws + WS_COUNTS;
  float* dw     = ws + WS_DW;
  float lossAcc = 0.0f;

  // Rows 0-7 of each 16-row tile live in lane 0's slots, rows 8-15 in lane 16's.
#pragma unroll
  for (int r = 0; r < 32; ++r) {
    int idx;
    if (r < 16) idx = __shfl(besti0[r & 7], (r < 8) ? 0 : 16, 32);
    else        idx = __shfl(besti1[r & 7], ((r & 15) < 8) ? 0 : 16, 32);
    const int row = r0 + r;
    const size_t zoff = (size_t)row * DDIM + lane;   // lane == feature dim
    const float zv = zf[zoff];
    const float qv = emb[(size_t)idx * DDIM + lane];
    outQ[zoff] = qv;                                  // quantized_st == quantized
    const float d = qv - zv;
    lossAcc = fmaf(d, d, lossAcc);
    atomicAdd(&dw[idx * DDIM + lane], zv);
    if (lane == 0) {
      outIdx[row] = (float)idx;
      atomicAdd(&counts[idx], 1.0f);
    }
  }

  // Wave-level sum of loss, one atomic per wave.
#pragma unroll
  for (int m = 16; m >= 1; m >>= 1) lossAcc += __shfl_xor(lossAcc, m, 32);
  if (lane == 0) atomicAdd(ws + WS_LOSS, lossAcc);
}

// ---------------------------------------------------------------------------
// Finalize: EMA update, Laplace-normalized cluster sizes, new embeddings, loss
// ---------------------------------------------------------------------------
__global__ __launch_bounds__(1024)
void vq_finalize_kernel(const float* __restrict__ ws,
                        const float* __restrict__ cluster_size,
                        const float* __restrict__ ema_in,
                        float* __restrict__ outLoss,
                        float* __restrict__ outEmb,
                        float* __restrict__ outNC,
                        float* __restrict__ outEMA,
                        float totalElems) {
  __shared__ float red[KCLUST];
  const int t = threadIdx.x;

  const float raw = cluster_size[t] * DECAYF + ws[WS_COUNTS + t] * OMDF;
  red[t] = raw;
  __syncthreads();
#pragma unroll 1
  for (int s = 512; s > 0; s >>= 1) {
    if (t < s) red[t] += red[t + s];
    __syncthreads();
  }
  const float n = red[0];

  const float newc = (raw + EPSF) / (n + (float)KCLUST * EPSF) * n;
  outNC[t] = newc;
  const float inv = 1.0f / newc;

#pragma unroll
  for (int d = 0; d < DDIM; ++d) {
    const int off = t * DDIM + d;
    const float e = ema_in[off] * DECAYF + ws[WS_DW + off] * OMDF;
    outEMA[off] = e;
    outEmb[off] = e * inv;
  }
  if (t == 0) outLoss[0] = ws[WS_LOSS] * (COMMITF / totalElems);
}

// ---------------------------------------------------------------------------
extern "C" void kernel_launch(void* const* d_in, const int* in_sizes, int n_in,
                              void* d_out, int out_size, void* d_ws, size_t ws_size,
                              hipStream_t stream) {
  const float* z   = (const float*)d_in[0];
  const float* emb = (const float*)d_in[1];
  const float* ema = (const float*)d_in[2];
  const float* csz = (const float*)d_in[3];
  float* out = (float*)d_out;
  float* ws  = (float*)d_ws;

  const int N = in_sizes[0] / DDIM;          // 262144 rows
  const size_t Q_OFF   = 0;
  const size_t IDX_OFF = (size_t)N * DDIM;   // 8388608
  const size_t L_OFF   = IDX_OFF + (size_t)N;
  const size_t EMB_OFF = L_OFF + 1;
  const size_t NC_OFF  = EMB_OFF + (size_t)KCLUST * DDIM;
  const size_t EMA_OFF = NC_OFF + KCLUST;

  // init: covers max(WS_ZERO_N, KCLUST) threads
  {
    int threads = WS_ZERO_N + 1024;          // safely covers both tasks
    int blocks = (threads + 255) / 256;
    vq_init_kernel<<<blocks, 256, 0, stream>>>(ws, emb);
  }

  // main: 32-row tiles, 8 waves per block
  {
    const int numRowTiles = (N + 31) / 32;
    const int blocks = (numRowTiles + 7) / 8;
    vq_main_kernel<<<blocks, 256, 0, stream>>>(z, emb, ws,
                                               out + Q_OFF, out + IDX_OFF,
                                               numRowTiles);
  }

  // finalize
  vq_finalize_kernel<<<1, 1024, 0, stream>>>(ws, csz, ema,
                                             out + L_OFF, out + EMB_OFF,
                                             out + NC_OFF, out + EMA_OFF,
                                             (float)((size_t)N * DDIM));
}